// STLT_45818711114029
// MI455X (gfx1250) — compile-verified
//
#include <hip/hip_runtime.h>
#include <math.h>

typedef __attribute__((ext_vector_type(2))) float v2f;
typedef __attribute__((ext_vector_type(8))) float v8f;

constexpr int NTOK = 65536;          // 16 * 64 * 64 tokens, dm = 256
constexpr int EPI_GELU  = 1;
constexpr int EPI_RESID = 2;

// ---------------------------------------------------------------------------
// WMMA fp32 GEMM:  C[M,N] = epi( A[M,K] @ W[N,K]^T + bias[N] (+ R[M,N]) )
// Tile 128x128, 8 waves, wave = 16 rows x 128 cols (8 v8f accumulators).
// Uses V_WMMA_F32_16X16X4_F32 (CDNA5 fp32 matrix pipe, exact fp32 inputs).
// ---------------------------------------------------------------------------
__global__ __launch_bounds__(256) void stlt_gemm_wmma(
    const float* __restrict__ A, const float* __restrict__ W,
    const float* __restrict__ bias, const float* __restrict__ R,
    float* __restrict__ C, int M, int N, int K, int flags)
{
  __shared__ __align__(16) float As[128][16];
  __shared__ __align__(16) float Ws[128][16];
  const int tid  = threadIdx.x;
  const int wave = tid >> 5;
  const int lane = tid & 31;
  const int ln   = lane & 15;
  const int hi   = lane >> 4;
  const int m0   = blockIdx.y * 128;
  const int n0   = blockIdx.x * 128;

  v8f acc[8] = {};

  const int lr = tid >> 1;        // 0..127 (tile row loaded by this thread)
  const int lk = (tid & 1) * 8;   // 0 or 8
  const float* aptr = A + (size_t)(m0 + lr) * K + lk;
  const float* wptr = W + (size_t)(n0 + lr) * K + lk;

  for (int k0 = 0; k0 < K; k0 += 16) {
    float4 a0 = *(const float4*)(aptr + k0);
    float4 a1 = *(const float4*)(aptr + k0 + 4);
    float4 w0 = *(const float4*)(wptr + k0);
    float4 w1 = *(const float4*)(wptr + k0 + 4);
    *(float4*)&As[lr][lk]     = a0;
    *(float4*)&As[lr][lk + 4] = a1;
    *(float4*)&Ws[lr][lk]     = w0;
    *(float4*)&Ws[lr][lk + 4] = w1;
    __syncthreads();
    const int ar = wave * 16 + ln;
#pragma unroll
    for (int kk = 0; kk < 4; ++kk) {
      // A fragment (16x4 fp32): lanes 0-15 hold K=0,1; lanes 16-31 hold K=2,3
      v2f a;
      a.x = As[ar][kk * 4 + 2 * hi];
      a.y = As[ar][kk * 4 + 2 * hi + 1];
#pragma unroll
      for (int t = 0; t < 8; ++t) {
        // B fragment (4x16): B[k][n] = W[n][k]
        v2f b;
        b.x = Ws[t * 16 + ln][kk * 4 + 2 * hi];
        b.y = Ws[t * 16 + ln][kk * 4 + 2 * hi + 1];
        acc[t] = __builtin_amdgcn_wmma_f32_16x16x4_f32(
            false, a, false, b, (short)0, acc[t], false, false);
      }
    }
    __syncthreads();
  }

  const int crow = m0 + wave * 16;
#pragma unroll
  for (int t = 0; t < 8; ++t) {
    const int n = n0 + t * 16 + ln;
    const float bv = bias ? bias[n] : 0.0f;
#pragma unroll
    for (int v = 0; v < 8; ++v) {
      const int m = crow + v + 8 * hi;   // C layout: vgpr v -> row v (lo) / v+8 (hi)
      float c = acc[t][v] + bv;
      if (flags & EPI_RESID) c += R[(size_t)m * N + n];
      if (flags & EPI_GELU)  c = 0.5f * c * (1.0f + erff(c * 0.70710678118654752f));
      C[(size_t)m * N + n] = c;
    }
  }
}

// ---------------------------------------------------------------------------
// Circular conv1d (kernel 3, padding 1, wrap), strided input view.
// y[n,o,l] = b[o] + sum_{c,k} x(n,c,(l+k-1) mod L) * w[o,c,k]
// ---------------------------------------------------------------------------
__global__ void stlt_conv1d_circ(
    const float* __restrict__ x, const float* __restrict__ w,
    const float* __restrict__ b, float* __restrict__ y,
    int Nb, int Cin, int L, int Cout, int sn, int sc, int sl)
{
  int idx = blockIdx.x * blockDim.x + threadIdx.x;
  int total = Nb * Cout * L;
  if (idx >= total) return;
  int l = idx % L;
  int o = (idx / L) % Cout;
  int n = idx / (L * Cout);
  int lm = (l + L - 1) % L, lp = (l + 1) % L;
  const float* xb = x + (size_t)n * sn;
  const float* wb = w + (size_t)o * Cin * 3;
  float acc = b[o];
  for (int c = 0; c < Cin; ++c) {
    const float* xc = xb + (size_t)c * sc;
    acc += wb[c * 3 + 0] * xc[(size_t)lm * sl]
         + wb[c * 3 + 1] * xc[(size_t)l  * sl]
         + wb[c * 3 + 2] * xc[(size_t)lp * sl];
  }
  y[idx] = acc;
}

// ---------------------------------------------------------------------------
// Token conv (Cin=1, circular, L=64) + transpose + sinusoidal PE, fused.
// out[n,l,d] = tb[d] + sum_k tw[d,k]*h[n,(l+k-1)%64] + PE(l,d)
// ---------------------------------------------------------------------------
__global__ void stlt_tok_embed(
    const float* __restrict__ h, const float* __restrict__ tw,
    const float* __restrict__ tb, float* __restrict__ out)
{
  int idx = blockIdx.x * blockDim.x + threadIdx.x;   // over 1024*64*256
  if (idx >= NTOK * 256) return;
  int d = idx & 255;
  int l = (idx >> 8) & 63;
  int n = idx >> 14;
  const float* hr = h + n * 64;
  int lm = (l + 63) & 63, lp = (l + 1) & 63;
  float v = tb[d] + tw[d * 3] * hr[lm] + tw[d * 3 + 1] * hr[l] + tw[d * 3 + 2] * hr[lp];
  int j = d >> 1;
  float f = expf(-(float)(2 * j) * (9.210340371976184f / 256.0f));  // 10000^(-2j/256)
  float ang = (float)l * f;
  v += (d & 1) ? cosf(ang) : sinf(ang);
  out[idx] = v;
}

// ---------------------------------------------------------------------------
// Fused pooled linear-attention: q/k viewed as (1024, 64, 8, 32).
// kp = mean-pool k over groups of 4 -> (16,32) in LDS; softmax(q.kp/sqrt(32));
// out = A @ kp written IN PLACE over Q. One block per (r, head), 64 threads.
// ---------------------------------------------------------------------------
__global__ __launch_bounds__(64) void stlt_pool_attn(
    float* __restrict__ Q, const float* __restrict__ Kk, int causal)
{
  __shared__ float kp[16][32];
  const int r = blockIdx.x;    // 0..1023
  const int h = blockIdx.y;    // 0..7
  const int t = threadIdx.x;   // 0..63
  for (int i = t; i < 512; i += 64) {
    int s = i >> 5, e = i & 31;
    size_t base = ((size_t)r * 64) * 256 + h * 32 + e;
    float acc = 0.0f;
#pragma unroll
    for (int j = 0; j < 4; ++j) acc += Kk[base + (size_t)(4 * s + j) * 256];
    kp[s][e] = 0.25f * acc;
  }
  __syncthreads();
  const int l = t;
  size_t qbase = ((size_t)(r * 64 + l)) * 256 + h * 32;
  float q[32];
#pragma unroll
  for (int e = 0; e < 32; ++e) q[e] = Q[qbase + e];
  const float inv = 0.17677669529663687f;  // 1/sqrt(32)
  float sc[16];
  float mx = -1e30f;
#pragma unroll
  for (int s = 0; s < 16; ++s) {
    float d = 0.0f;
#pragma unroll
    for (int e = 0; e < 32; ++e) d += q[e] * kp[s][e];
    d *= inv;
    if (causal && s > l) d = -1e30f;
    sc[s] = d;
    if (d > mx) mx = d;
  }
  float sum = 0.0f;
#pragma unroll
  for (int s = 0; s < 16; ++s) { float e_ = expf(sc[s] - mx); sc[s] = e_; sum += e_; }
  float rs = 1.0f / sum;
  float out[32];
#pragma unroll
  for (int e = 0; e < 32; ++e) out[e] = 0.0f;
#pragma unroll
  for (int s = 0; s < 16; ++s) {
    float a = sc[s] * rs;
#pragma unroll
    for (int e = 0; e < 32; ++e) out[e] += a * kp[s][e];
  }
#pragma unroll
  for (int e = 0; e < 32; ++e) Q[qbase + e] = out[e];
}

// ---------------------------------------------------------------------------
// LayerNorm over 256, one wave32 per token (8 tokens / block).
// out = LN(a [+ b2]) * w + bias.  Safe for out==a (in place).
// ---------------------------------------------------------------------------
__global__ __launch_bounds__(256) void stlt_layernorm(
    float* __restrict__ out, const float* __restrict__ a,
    const float* __restrict__ b2, const float* __restrict__ w,
    const float* __restrict__ bias, int ntok)
{
  int tok = blockIdx.x * 8 + (threadIdx.x >> 5);
  if (tok >= ntok) return;
  int lane = threadIdx.x & 31;
  const float* ar = a + (size_t)tok * 256;
  const float* br = b2 ? b2 + (size_t)tok * 256 : nullptr;
  float vals[8];
  float sum = 0.0f, sq = 0.0f;
#pragma unroll
  for (int i = 0; i < 8; ++i) {
    float v = ar[i * 32 + lane];
    if (br) v += br[i * 32 + lane];
    vals[i] = v; sum += v; sq += v * v;
  }
#pragma unroll
  for (int o = 16; o > 0; o >>= 1) {
    sum += __shfl_xor(sum, o, 32);
    sq  += __shfl_xor(sq,  o, 32);
  }
  float mean = sum * (1.0f / 256.0f);
  float var  = sq * (1.0f / 256.0f) - mean * mean;
  float rstd = rsqrtf(var + 1e-5f);
  float* orow = out + (size_t)tok * 256;
#pragma unroll
  for (int i = 0; i < 8; ++i) {
    int d = i * 32 + lane;
    orow[d] = (vals[i] - mean) * rstd * w[d] + bias[d];
  }
}

// Wo (256,512) -> W2[n,k] = Wo[n,k] + Wo[n,k+256]  (out_t == out_s fold)
__global__ void stlt_wsum(const float* __restrict__ Wo, float* __restrict__ W2)
{
  int idx = blockIdx.x * blockDim.x + threadIdx.x;
  if (idx >= 256 * 256) return;
  int n = idx >> 8, k = idx & 255;
  W2[idx] = Wo[n * 512 + k] + Wo[n * 512 + 256 + k];
}

// mean over s axis: dec (16,64,64,256) -> (16,64,256)
__global__ void stlt_mean_s(const float* __restrict__ dec, float* __restrict__ out)
{
  int bt = blockIdx.x;          // 0..1023
  int d  = threadIdx.x;         // 0..255
  size_t base = (size_t)bt * 64 * 256 + d;
  float acc = 0.0f;
  for (int s = 0; s < 64; ++s) acc += dec[base + (size_t)s * 256];
  out[bt * 256 + d] = acc * (1.0f / 64.0f);
}

// final proj: out[b, t(0..31), o] = mean[b, t+32, :] . proj_w[o,:] + proj_b[o]
__global__ void stlt_final_proj(
    const float* __restrict__ mean, const float* __restrict__ pw,
    const float* __restrict__ pb, float* __restrict__ out)
{
  int idx = blockIdx.x * blockDim.x + threadIdx.x;   // 16*32*64
  if (idx >= 16 * 32 * 64) return;
  int o = idx & 63;
  int t = (idx >> 6) & 31;
  int b = idx >> 11;
  const float* mr = mean + ((size_t)b * 64 + (t + 32)) * 256;
  const float* wr = pw + o * 256;
  float acc = pb[o];
  for (int d = 0; d < 256; ++d) acc += mr[d] * wr[d];
  out[idx] = acc;
}

// ---------------------------------------------------------------------------
extern "C" void kernel_launch(void* const* d_in, const int* in_sizes, int n_in,
                              void* d_out, int out_size, void* d_ws, size_t ws_size,
                              hipStream_t stream)
{
  (void)in_sizes; (void)n_in; (void)out_size; (void)ws_size;
  auto F = [&](int i) { return (const float*)d_in[i]; };
  const float *x_enc = F(0), *x_dec = F(1);
  const float *enc_spw = F(2), *enc_spb = F(3), *enc_scw = F(4), *enc_scb = F(5),
              *enc_tw = F(6), *enc_tb = F(7);
  const float *dec_spw = F(8), *dec_spb = F(9), *dec_scw = F(10), *dec_scb = F(11),
              *dec_tw = F(12), *dec_tb = F(13);
  const float *e_Wq = F(14), *e_bq = F(15), *e_Wo = F(16), *e_bo = F(17);
  const float *e_c1w = F(18), *e_c1b = F(19), *e_c2w = F(20), *e_c2b = F(21);
  const float *e_ln1w = F(22), *e_ln1b = F(23), *e_ln2w = F(24), *e_ln2b = F(25);
  const float *enc_nw = F(26), *enc_nb = F(27);
  const float *d_sWq = F(28), *d_sbq = F(29), *d_sWo = F(30), *d_sbo = F(31);
  const float *d_cWq = F(32), *d_cbq = F(33), *d_cWo = F(34), *d_cbo = F(35);
  const float *d_c1w = F(36), *d_c1b = F(37), *d_c2w = F(38), *d_c2b = F(39);
  const float *d_ln1w = F(40), *d_ln1b = F(41), *d_ln2w = F(42), *d_ln2b = F(43),
              *d_ln3w = F(44), *d_ln3b = F(45);
  const float *dec_nw = F(46), *dec_nb = F(47), *proj_w = F(48), *proj_b = F(49);

  float* ws = (float*)d_ws;
  const size_t BIG = (size_t)NTOK * 256;     // 16M floats = 64 MiB
  float* A_ENC = ws;                          // encoder activations
  float* A_DEC = ws + BIG;                    // decoder activations
  float* T1    = ws + 2 * BIG;                // proj / attn-out / FFN hidden chunk
  float* T2    = ws + 3 * BIG;                // cross-K proj / FFN output / emb temps
  float* W2    = ws + 4 * BIG;                // 64K floats
  float* MEANB = W2 + 256 * 256;              // 1024*256 floats

  auto gemm = [&](const float* A, const float* Wm, const float* bias, const float* R,
                  float* C, int M, int N, int K, int flags) {
    dim3 g(N / 128, M / 128);
    stlt_gemm_wmma<<<g, dim3(256), 0, stream>>>(A, Wm, bias, R, C, M, N, K, flags);
  };
  auto ln = [&](float* out, const float* a, const float* b2,
                const float* w, const float* bias) {
    stlt_layernorm<<<NTOK / 8, 256, 0, stream>>>(out, a, b2, w, bias, NTOK);
  };
  auto attn = [&](float* Qbuf, const float* Kbuf, int causal) {
    stlt_pool_attn<<<dim3(1024, 8), 64, 0, stream>>>(Qbuf, Kbuf, causal);
  };
  auto ffn = [&](float* Xh, const float* c1w_, const float* c1b_,
                 const float* c2w_, const float* c2b_) {
    for (int c0 = 0; c0 < NTOK; c0 += 8192) {
      gemm(Xh + (size_t)c0 * 256, c1w_, c1b_, nullptr, T1, 8192, 1024, 256, EPI_GELU);
      gemm(T1, c2w_, c2b_, nullptr, T2 + (size_t)c0 * 256, 8192, 256, 1024, 0);
    }
  };
  auto embed = [&](const float* x, const float* spw, const float* spb,
                   const float* scw, const float* scb,
                   const float* tw, const float* tb, float* out) {
    float* h1 = T2;                         // (16,96,96)
    float* h2 = h1 + 16 * 96 * 96;          // (16,64,96)
    float* h3 = h2 + 16 * 64 * 96;          // (16,64,64)
    // conv over swapped x: (B, C=64, L=96), x[n,l,c] -> sn=6144, sc=1, sl=64
    stlt_conv1d_circ<<<(16 * 96 * 96 + 255) / 256, 256, 0, stream>>>(
        x, spw, spb, h1, 16, 64, 96, 96, 6144, 1, 64);
    // conv: (B, C=96, L=96) contiguous
    stlt_conv1d_circ<<<(16 * 64 * 96 + 255) / 256, 256, 0, stream>>>(
        h1, scw, scb, h2, 16, 96, 96, 64, 9216, 96, 1);
    // conv over swapped h2: (B, C=96, L=64), h2[n,l,c] -> sn=6144, sc=1, sl=96
    stlt_conv1d_circ<<<(16 * 64 * 64 + 255) / 256, 256, 0, stream>>>(
        h2, scw, scb, h3, 16, 96, 64, 64, 6144, 1, 96);
    stlt_tok_embed<<<(NTOK * 256 + 255) / 256, 256, 0, stream>>>(h3, tw, tb, out);
  };

  // ---------------- Encoder ----------------
  embed(x_enc, enc_spw, enc_spb, enc_scw, enc_scb, enc_tw, enc_tb, A_ENC);
  for (int i = 0; i < 3; ++i) {
    const float* Wq = e_Wq + (size_t)i * 256 * 256;
    gemm(A_ENC, Wq, e_bq + i * 256, nullptr, T1, NTOK, 256, 256, 0);     // P=q=k
    attn(T1, T1, 0);                                                      // Y in place
    stlt_wsum<<<256, 256, 0, stream>>>(e_Wo + (size_t)i * 256 * 512, W2);
    gemm(T1, W2, e_bo + i * 256, A_ENC, A_ENC, NTOK, 256, 256, EPI_RESID); // h = x+attn
    ln(A_ENC, A_ENC, nullptr, e_ln1w + i * 256, e_ln1b + i * 256);
    ffn(A_ENC, e_c1w + (size_t)i * 1024 * 256, e_c1b + i * 1024,
        e_c2w + (size_t)i * 256 * 1024, e_c2b + i * 256);
    ln(A_ENC, A_ENC, T2, e_ln2w + i * 256, e_ln2b + i * 256);            // LN(h+ffn)
  }
  ln(A_ENC, A_ENC, nullptr, enc_nw, enc_nb);

  // ---------------- Decoder ----------------
  embed(x_dec, dec_spw, dec_spb, dec_scw, dec_scb, dec_tw, dec_tb, A_DEC);
  for (int i = 0; i < 2; ++i) {
    // self-attention (causal)
    gemm(A_DEC, d_sWq + (size_t)i * 256 * 256, d_sbq + i * 256, nullptr,
         T1, NTOK, 256, 256, 0);
    attn(T1, T1, 1);
    stlt_wsum<<<256, 256, 0, stream>>>(d_sWo + (size_t)i * 256 * 512, W2);
    gemm(T1, W2, d_sbo + i * 256, A_DEC, A_DEC, NTOK, 256, 256, EPI_RESID);
    ln(A_DEC, A_DEC, nullptr, d_ln1w + i * 256, d_ln1b + i * 256);
    // cross-attention (kv = encoder output)
    gemm(A_DEC, d_cWq + (size_t)i * 256 * 256, d_cbq + i * 256, nullptr,
         T1, NTOK, 256, 256, 0);
    gemm(A_ENC, d_cWq + (size_t)i * 256 * 256, d_cbq + i * 256, nullptr,
         T2, NTOK, 256, 256, 0);
    attn(T1, T2, 0);
    stlt_wsum<<<256, 256, 0, stream>>>(d_cWo + (size_t)i * 256 * 512, W2);
    gemm(T1, W2, d_cbo + i * 256, A_DEC, A_DEC, NTOK, 256, 256, EPI_RESID);
    ln(A_DEC, A_DEC, nullptr, d_ln2w + i * 256, d_ln2b + i * 256);
    // FFN
    ffn(A_DEC, d_c1w + (size_t)i * 1024 * 256, d_c1b + i * 1024,
        d_c2w + (size_t)i * 256 * 1024, d_c2b + i * 256);
    ln(A_DEC, A_DEC, T2, d_ln3w + i * 256, d_ln3b + i * 256);
  }
  ln(A_DEC, A_DEC, nullptr, dec_nw, dec_nb);

  // ---------------- Head ----------------
  stlt_mean_s<<<1024, 256, 0, stream>>>(A_DEC, MEANB);
  stlt_final_proj<<<(16 * 32 * 64 + 255) / 256, 256, 0, stream>>>(
      MEANB, proj_w, proj_b, (float*)d_out);
}